// APPNPModel_14594298872378
// MI455X (gfx1250) — compile-verified
//
#include <hip/hip_runtime.h>

#define N_NODES 100000
#define N_EDGES 3200000
#define FDIM    512
#define HID     64
#define ALPHA   0.1f
#define ITERS   10

typedef __attribute__((ext_vector_type(2))) float v2f;
typedef __attribute__((ext_vector_type(8))) float v8f;

// Guaranteed single-instruction fp32 add atomic at device scope (L2 RMW).
// No-return form -> tracked on STOREcnt, no data round-trip to VGPRs.
__device__ __forceinline__ void atomic_add_f32_dev(float* p, float x) {
    asm volatile("global_atomic_add_f32 %0, %1, off scope:SCOPE_DEV"
                 :: "v"(p), "v"(x)
                 : "memory");
}

// SpMM scatter: acc[row] += w * p[col].  16 threads per edge, float4 per thread.
// p (25.6 MB) is L2-resident; gather is a coalesced 256B burst per edge;
// scatter is hardware global_atomic_add_f32 into the L2 atomic units.
__global__ __launch_bounds__(256) void appnp_spmm(
    const int* __restrict__ ei, const float* __restrict__ ew,
    const float* __restrict__ p, float* __restrict__ acc)
{
    long long g = (long long)blockIdx.x * 256 + threadIdx.x;
    long long e = g >> 4;
    int chunk   = (int)(g & 15);
    if (e >= N_EDGES) return;
    const int   r = ei[e];
    const int   c = ei[(long long)N_EDGES + e];
    const float w = ew[e];
    const float4 v = *reinterpret_cast<const float4*>(p + (long long)c * HID + chunk * 4);
    float* dst = acc + (long long)r * HID + chunk * 4;
    atomic_add_f32_dev(dst + 0, w * v.x);
    atomic_add_f32_dev(dst + 1, w * v.y);
    atomic_add_f32_dev(dst + 2, w * v.z);
    atomic_add_f32_dev(dst + 3, w * v.w);
}

// D(16x16,f32) = A(16x4,f32) x B(4x16,f32) + C   -- CDNA5 V_WMMA_F32_16X16X4_F32
__device__ __forceinline__ v8f wmma_f32_16x16x4(v2f a, v2f b, v8f c) {
    return __builtin_amdgcn_wmma_f32_16x16x4_f32(
        /*neg_a=*/false, a, /*neg_b=*/false, b,
        /*c_mod=*/(short)0, c, /*reuse_a=*/false, /*reuse_b=*/false);
}

// Fused MLP head: z = relu(F@W1 + b1) @ W2 + b2 ; also p = z (initial state).
// Block = 128 threads = 4 waves; block owns 16 node rows; wave w owns cols 16w..16w+15.
__global__ __launch_bounds__(128) void appnp_mlp(
    const float* __restrict__ feat, const float* __restrict__ W1,
    const float* __restrict__ b1,   const float* __restrict__ W2,
    const float* __restrict__ b2,   float* __restrict__ z, float* __restrict__ p)
{
    constexpr int FSTR = 516;  // 516%64==4 -> float2 A reads hit banks 4m+k..4m+k+3: all 64 banks
    constexpr int HSTR = 68;   // same property for the 16x64 h tile
    __shared__ float ftile[16 * FSTR];
    __shared__ float htile[16 * HSTR];

    const int tid   = threadIdx.x;
    const int lane  = tid & 31;
    const int wave  = tid >> 5;
    const int r0    = blockIdx.x * 16;        // 100000 = 6250 * 16, no remainder
    const int c0    = wave * 16;
    const int n15   = lane & 15;              // A row (M) and B col (N) index
    const int klo   = (lane >> 4) << 1;       // lanes 16..31 hold K+2,K+3 (ISA 7.12.2)
    const int mbase = (lane >> 4) << 3;       // C/D: lanes 16..31 hold M+8

    // Stage 16x512 feature tile into LDS, fully coalesced float4 global loads.
    #pragma unroll
    for (int i = 0; i < 16; ++i) {
        int idx = i * 128 + tid;              // 2048 float4 total
        int r   = idx >> 7;
        int cq  = idx & 127;
        const float4 v = *reinterpret_cast<const float4*>(
            feat + (long long)(r0 + r) * FDIM + cq * 4);
        float* dst = &ftile[r * FSTR + cq * 4];
        dst[0] = v.x; dst[1] = v.y; dst[2] = v.z; dst[3] = v.w;
    }
    __syncthreads();

    // GEMM1: K=512 in steps of 4 -> 128 WMMA per wave.
    v8f acc = {};
    #pragma unroll 4
    for (int k = 0; k < FDIM; k += 4) {
        v2f a = *reinterpret_cast<const v2f*>(&ftile[n15 * FSTR + k + klo]);
        v2f b;
        b.x = W1[(k + klo) * HID + c0 + n15];
        b.y = W1[(k + klo + 1) * HID + c0 + n15];
        acc = wmma_f32_16x16x4(a, b, acc);
    }

    // bias + ReLU, scatter tile into LDS h-tile (rows of h, all 64 cols per block).
    const float bb1 = b1[c0 + n15];
    #pragma unroll
    for (int v = 0; v < 8; ++v) {
        float hv = acc[v] + bb1;
        hv = hv > 0.f ? hv : 0.f;
        htile[(mbase + v) * HSTR + c0 + n15] = hv;
    }
    __syncthreads();

    // GEMM2: K=64 -> 16 WMMA per wave.
    v8f acc2 = {};
    #pragma unroll
    for (int k = 0; k < HID; k += 4) {
        v2f a = *reinterpret_cast<const v2f*>(&htile[n15 * HSTR + k + klo]);
        v2f b;
        b.x = W2[(k + klo) * HID + c0 + n15];
        b.y = W2[(k + klo + 1) * HID + c0 + n15];
        acc2 = wmma_f32_16x16x4(a, b, acc2);
    }
    const float bb2 = b2[c0 + n15];
    #pragma unroll
    for (int v = 0; v < 8; ++v) {
        float zv = acc2[v] + bb2;
        long long o = (long long)(r0 + mbase + v) * HID + (c0 + n15);
        z[o] = zv;
        p[o] = zv;   // initial propagation state
    }
}

// Zero the accumulator (float4 stores).
__global__ __launch_bounds__(256) void appnp_zero(float* __restrict__ acc)
{
    int i = blockIdx.x * 256 + threadIdx.x;          // 1.6M float4
    float4 zero; zero.x = zero.y = zero.z = zero.w = 0.f;
    reinterpret_cast<float4*>(acc)[i] = zero;
}

// p = (1-alpha)*acc + alpha*z ; acc = 0 for the next iteration.
__global__ __launch_bounds__(256) void appnp_combine(
    float* __restrict__ acc, const float* __restrict__ z, float* __restrict__ p)
{
    int i = blockIdx.x * 256 + threadIdx.x;          // 1.6M float4
    const float4 a  = reinterpret_cast<const float4*>(acc)[i];
    const float4 zz = reinterpret_cast<const float4*>(z)[i];
    float4 o;
    o.x = (1.f - ALPHA) * a.x + ALPHA * zz.x;
    o.y = (1.f - ALPHA) * a.y + ALPHA * zz.y;
    o.z = (1.f - ALPHA) * a.z + ALPHA * zz.z;
    o.w = (1.f - ALPHA) * a.w + ALPHA * zz.w;
    reinterpret_cast<float4*>(p)[i] = o;
    float4 zero; zero.x = zero.y = zero.z = zero.w = 0.f;
    reinterpret_cast<float4*>(acc)[i] = zero;
}

extern "C" void kernel_launch(void* const* d_in, const int* in_sizes, int n_in,
                              void* d_out, int out_size, void* d_ws, size_t ws_size,
                              hipStream_t stream) {
    (void)in_sizes; (void)n_in; (void)out_size; (void)ws_size;
    const float* feat = (const float*)d_in[0];
    const float* W1   = (const float*)d_in[1];
    const float* b1   = (const float*)d_in[2];
    const float* W2   = (const float*)d_in[3];
    const float* b2   = (const float*)d_in[4];
    const float* ew   = (const float*)d_in[5];
    const int*   ei   = (const int*)d_in[6];

    float* z   = (float*)d_ws;                              // 25.6 MB
    float* acc = z + (size_t)N_NODES * HID;                 // 25.6 MB
    float* p   = (float*)d_out;                             // propagation state

    const int vec_blocks  = (N_NODES * HID / 4) / 256;      // 6250
    const int spmm_blocks = (int)(((long long)N_EDGES * 16) / 256);  // 200000

    appnp_mlp<<<N_NODES / 16, 128, 0, stream>>>(feat, W1, b1, W2, b2, z, p);
    appnp_zero<<<vec_blocks, 256, 0, stream>>>(acc);
    for (int it = 0; it < ITERS; ++it) {
        appnp_spmm<<<spmm_blocks, 256, 0, stream>>>(ei, ew, p, acc);
        appnp_combine<<<vec_blocks, 256, 0, stream>>>(acc, z, p);
    }
}